// Attn_86560770884125
// MI455X (gfx1250) — compile-verified
//
#include <hip/hip_runtime.h>
#include <hip/hip_bf16.h>

typedef __attribute__((ext_vector_type(2))) float v2f;
typedef __attribute__((ext_vector_type(4))) float v4f;
typedef __attribute__((ext_vector_type(8))) float v8f;

constexpr int Bb = 64;    // batch
constexpr int Ll = 1024;  // encoder length
constexpr int Hh = 1024;  // hidden dim

// ---------------------------------------------------------------------------
// Kernel 1: u[b,h] = sum_g hidden[b,g] * W[g,h]   (64x1024 @ 1024x1024 GEMM)
// One wave per 16x16 C tile, K-loop in steps of 4 via v_wmma_f32_16x16x4_f32.
// 4 M-tiles x 64 N-tiles = 256 tiles = 32 blocks x 8 waves.
// ---------------------------------------------------------------------------
__global__ __launch_bounds__(256) void Attn_proj_wmma_kernel(
    const float* __restrict__ hid,   // [64,1024] row-major
    const float* __restrict__ W,     // [1024(g),1024(h)] row-major
    float* __restrict__ u)           // [64,1024]
{
    const int wave = threadIdx.x >> 5;
    const int lane = threadIdx.x & 31;
    const int tile = blockIdx.x * 8 + wave;          // 0..255
    const int m0 = (tile & 3) * 16;                  // b-tile
    const int n0 = (tile >> 2) * 16;                 // h-tile

    const int lm = lane & 15;                        // M (A) / N (B) within tile
    const int g2 = (lane >> 4) * 2;                  // K sub-offset per lane half

    const float* Arow = hid + (size_t)(m0 + lm) * Hh;   // + k
    const float* Bcol = W + (n0 + lm);                  // + k*Hh

    v8f c = {};
#pragma unroll 4
    for (int k = 0; k < Hh; k += 4) {
        v2f a, bb;
        // A 16x4 f32: VGPR v holds K = k + g2 + v (contiguous pair per lane)
        a.x = Arow[k + g2];
        a.y = Arow[k + g2 + 1];
        // B 4x16 f32: VGPR v holds K = k + g2 + v, N = lane&15
        bb.x = Bcol[(size_t)(k + g2) * Hh];
        bb.y = Bcol[(size_t)(k + g2 + 1) * Hh];
        c = __builtin_amdgcn_wmma_f32_16x16x4_f32(
            /*neg_a=*/false, a, /*neg_b=*/false, bb,
            /*c_mod=*/(short)0, c, /*reuse_a=*/false, /*reuse_b=*/false);
    }

    // C 16x16 f32: VGPR r -> M = m0 + r + 8*(lane>>4), N = n0 + (lane&15)
    const int rbase = m0 + ((lane >> 4) << 3);
#pragma unroll
    for (int r = 0; r < 8; ++r)
        u[(size_t)(rbase + r) * Hh + (n0 + lm)] = c[r];
}

// ---------------------------------------------------------------------------
// Kernel 2: energies[b,l] = sum_h enc[l,b,h] * u[b,h]
// Bandwidth-bound: streams enc (256 MB) once with non-temporal b128 loads.
// Block = (b, 8 consecutive l); u[b,:] staged in LDS (4 KB); one wave per l;
// each lane reads float4 coalesced, then wave32 shfl_xor reduction.
// Writes energies directly into d_out (softmax runs in-place afterwards).
// ---------------------------------------------------------------------------
__global__ __launch_bounds__(256) void Attn_energies_kernel(
    const float* __restrict__ enc,   // [L,B,H]
    const float* __restrict__ u,     // [B,H]
    float* __restrict__ e)           // [B,L] (== d_out)
{
    __shared__ v4f us[Hh / 4];       // u[b,:] as float4, 4 KB

    const int b = blockIdx.y;
    const int l = blockIdx.x * 8 + (threadIdx.x >> 5);
    const int lane = threadIdx.x & 31;

    const v4f* urow = (const v4f*)(u + (size_t)b * Hh);
    for (int i = threadIdx.x; i < Hh / 4; i += blockDim.x)
        us[i] = urow[i];
    __syncthreads();

    const v4f* erow = (const v4f*)(enc + ((size_t)l * Bb + b) * Hh);
    float sum = 0.f;
#pragma unroll
    for (int i = 0; i < 8; ++i) {
        const int idx = lane + i * 32;               // coalesced 512B/step
        v4f ev = __builtin_nontemporal_load(&erow[idx]);  // read-once: NT hint
        v4f uv = us[idx];
        sum += ev.x * uv.x + ev.y * uv.y + ev.z * uv.z + ev.w * uv.w;
    }
#pragma unroll
    for (int off = 16; off; off >>= 1)
        sum += __shfl_xor(sum, off, 32);
    if (lane == 0)
        e[(size_t)b * Ll + l] = sum;
}

// ---------------------------------------------------------------------------
// Kernel 3: in-place row softmax over L for each b. One block per b.
// ---------------------------------------------------------------------------
__global__ __launch_bounds__(256) void Attn_softmax_kernel(float* __restrict__ out)
{
    __shared__ float red[8];
    const int b = blockIdx.x;
    float* row = out + (size_t)b * Ll;
    const int lane = threadIdx.x & 31;
    const int wave = threadIdx.x >> 5;

    float m = -3.402823466e38f;
    for (int i = threadIdx.x; i < Ll; i += 256)
        m = fmaxf(m, row[i]);
#pragma unroll
    for (int off = 16; off; off >>= 1)
        m = fmaxf(m, __shfl_xor(m, off, 32));
    if (lane == 0) red[wave] = m;
    __syncthreads();
    float mm = red[0];
#pragma unroll
    for (int j = 1; j < 8; ++j) mm = fmaxf(mm, red[j]);
    __syncthreads();

    float s = 0.f;
    for (int i = threadIdx.x; i < Ll; i += 256) {
        float v = __expf(row[i] - mm);
        row[i] = v;
        s += v;
    }
#pragma unroll
    for (int off = 16; off; off >>= 1)
        s += __shfl_xor(s, off, 32);
    if (lane == 0) red[wave] = s;
    __syncthreads();
    float tot = 0.f;
#pragma unroll
    for (int j = 0; j < 8; ++j) tot += red[j];
    const float inv = 1.0f / tot;
    for (int i = threadIdx.x; i < Ll; i += 256)
        row[i] *= inv;
}

extern "C" void kernel_launch(void* const* d_in, const int* in_sizes, int n_in,
                              void* d_out, int out_size, void* d_ws, size_t ws_size,
                              hipStream_t stream) {
    (void)in_sizes; (void)n_in; (void)out_size; (void)ws_size;
    const float* hidden = (const float*)d_in[0];   // (1,B,H)
    const float* enc    = (const float*)d_in[1];   // (L,B,H)
    const float* W      = (const float*)d_in[2];   // (H,H)
    // d_in[3] = bias: mathematically eliminated (softmax shift invariance).
    float* out = (float*)d_out;                    // (B,1,L) flat = b*L + l
    float* u   = (float*)d_ws;                     // 64*1024 floats = 256 KB

    Attn_proj_wmma_kernel<<<dim3(32), dim3(256), 0, stream>>>(hidden, W, u);
    Attn_energies_kernel<<<dim3(Ll / 8, Bb), dim3(256), 0, stream>>>(enc, u, out);
    Attn_softmax_kernel<<<dim3(Bb), dim3(256), 0, stream>>>(out);
}